// YoloV3LossV2_56925496541418
// MI455X (gfx1250) — compile-verified
//
#include <hip/hip_runtime.h>
#include <math.h>

#define NCLS   80
#define BATCH  16
#define NT     50
#define EPSF   1e-9f
#define FOUR_OVER_PI2 0.40528473456935108f

__device__ __constant__ float c_anchors[9][2] = {
    {10,13},{16,30},{33,23},{30,61},{62,45},{59,119},{116,90},{156,198},{373,326}};

// slots: [0,64) p3 conf, [64,96) p4 conf, [96,112) p5 conf, [112,160) cell losses
#define SLOT_P3   0
#define SLOT_P4   64
#define SLOT_P5   96
#define SLOT_CELL 112
#define NPART     160

typedef float v2f __attribute__((ext_vector_type(2)));
typedef float v8f __attribute__((ext_vector_type(8)));

__device__ __forceinline__ float sigmoidf_(float x) { return 1.0f / (1.0f + expf(-x)); }

// exact port of ciou_xywh (float32, EPS=1e-9, note h gets +EPS but w does not)
__device__ float ciou_xywh_dev(float cx1, float cy1, float w1i, float h1i,
                               float cx2, float cy2, float w2i, float h2i) {
    float b1x1 = cx1 - w1i * 0.5f, b1x2 = cx1 + w1i * 0.5f;
    float b1y1 = cy1 - h1i * 0.5f, b1y2 = cy1 + h1i * 0.5f;
    float b2x1 = cx2 - w2i * 0.5f, b2x2 = cx2 + w2i * 0.5f;
    float b2y1 = cy2 - h2i * 0.5f, b2y2 = cy2 + h2i * 0.5f;
    float iw = fminf(b1x2, b2x2) - fmaxf(b1x1, b2x1); iw = iw < 0.f ? 0.f : iw;
    float ih = fminf(b1y2, b2y2) - fmaxf(b1y1, b2y1); ih = ih < 0.f ? 0.f : ih;
    float inter = iw * ih;
    float w1 = b1x2 - b1x1, h1 = b1y2 - b1y1 + EPSF;
    float w2 = b2x2 - b2x1, h2 = b2y2 - b2y1 + EPSF;
    float uni = w1 * h1 + w2 * h2 - inter + EPSF;
    float iou = inter / uni;
    float cw = fmaxf(b1x2, b2x2) - fminf(b1x1, b2x1);
    float ch = fmaxf(b1y2, b2y2) - fminf(b1y1, b2y1);
    float c2 = cw * cw + ch * ch + EPSF;
    float dx = b2x1 + b2x2 - b1x1 - b1x2;
    float dy = b2y1 + b2y2 - b1y1 - b1y2;
    float rho2 = (dx * dx + dy * dy) * 0.25f;
    float d = atanf(w2 / h2) - atanf(w1 / h1);
    float v = FOUR_OVER_PI2 * d * d;
    float alpha = v / (v - iou + (1.0f + EPSF));
    return iou - (rho2 / c2 + v * alpha);
}

// Dense part of noobj loss: sum of sigmoid(conf)^2 over all (b, anchor, pix).
// conf channel of anchor a is plane (a*85+4) of the 255-plane stack.
__global__ __launch_bounds__(256)
void conf_sq_sum_kernel(const float* __restrict__ P, int HW, int nElems,
                        float* __restrict__ partials, int slotBase) {
    __shared__ float red[256];
    float acc = 0.0f;
    const int stride = gridDim.x * 256;
    for (int e = blockIdx.x * 256 + threadIdx.x; e < nElems; e += stride) {
        int b   = e / (3 * HW);
        int r   = e - b * 3 * HW;
        int a   = r / HW;
        int pix = r - a * HW;
        const float* addr = P + (size_t)(b * 255 + a * 85 + 4) * (size_t)HW + pix;
        __builtin_prefetch(addr + stride, 0, 0);   // global_prefetch_b8
        float pc = sigmoidf_(*addr);
        acc += pc * pc;
    }
    red[threadIdx.x] = acc;
    __syncthreads();
    for (int s = 128; s > 0; s >>= 1) {
        if (threadIdx.x < s) red[threadIdx.x] += red[threadIdx.x + s];
        __syncthreads();
    }
    if (threadIdx.x == 0) partials[slotBase + blockIdx.x] = red[0];
}

// One block per (layer, batch): encode targets (serial-scan semantics),
// then sparse gathers for box/obj/cls losses and noobj ignore corrections.
__global__ __launch_bounds__(256)
void cell_loss_kernel(const float* __restrict__ p3, const float* __restrict__ p4,
                      const float* __restrict__ p5, const float* __restrict__ target,
                      float* __restrict__ partials) {
    const int L = blockIdx.x / BATCH;
    const int b = blockIdx.x % BATCH;
    const float* P = (L == 0) ? p3 : (L == 1) ? p4 : p5;
    const int W  = (L == 0) ? 76 : (L == 1) ? 38 : 19;
    const int H  = W;
    const int HW = W * H;
    const float Wf = (float)W, Hf = (float)H;

    __shared__ int   s_ok[NT], s_bn[NT], s_gi[NT], s_gj[NT], s_ign[NT], s_cid[NT];
    __shared__ float s_tb[NT][4];
    __shared__ int      s_ckey[NT];
    __shared__ unsigned s_cmask[NT][3];
    __shared__ float    s_ctb[NT][4];
    __shared__ int   s_zkey[3 * NT];
    __shared__ int   s_nc, s_nz;
    __shared__ float red[256];

    // phase 1: per-target encode (vectorized over t)
    int t = threadIdx.x;
    if (t < NT) {
        const float* tg = target + ((size_t)b * NT + t) * 5;
        float cx = tg[0], cy = tg[1], w = tg[2], h = tg[3], cl = tg[4];
        bool valid = (cx + cy + w + h + cl) > 0.0f;
        float gwa = w * 608.0f, gha = h * 608.0f;
        float ious[9];
        float best = -1.0f; int bk = 0;
        for (int k = 0; k < 9; k++) {
            float aw = c_anchors[k][0], ah = c_anchors[k][1];
            float inter = fminf(gwa, aw) * fminf(gha, ah);
            float iou = inter / (gwa * gha + aw * ah - inter + 1e-16f);
            ious[k] = iou;
            if (iou > best) { best = iou; bk = k; }   // first max wins (argmax)
        }
        bool ok = valid && (bk >= 3 * L) && (bk < 3 * L + 3);
        int bn = bk - 3 * L; bn = bn < 0 ? 0 : (bn > 2 ? 2 : bn);
        float gx = cx * Wf, gy = cy * Hf, gw = w * Wf, gh = h * Hf;
        int gi = (int)gx, gj = (int)gy;
        int ign = 0;
        for (int a = 0; a < 3; a++) if (ious[3 * L + a] > 0.5f) ign |= (1 << a);
        float saw = c_anchors[3 * L + bn][0] * (Wf / 608.0f);
        float sah = c_anchors[3 * L + bn][1] * (Hf / 608.0f);
        int cid = (int)cl; cid = cid < 0 ? 0 : (cid > NCLS - 1 ? NCLS - 1 : cid);
        s_ok[t] = ok ? 1 : 0; s_bn[t] = bn; s_gi[t] = gi; s_gj[t] = gj;
        s_ign[t] = ign; s_cid[t] = cid;
        s_tb[t][0] = gx - (float)gi; s_tb[t][1] = gy - (float)gj;
        s_tb[t][2] = gw / saw;       s_tb[t][3] = gh / sah;
    }
    __syncthreads();

    // phase 2: serial winner resolution (matches jax.lax.scan write ordering):
    // tbox/mask/tconf: last writer wins; tcls: bits accumulate (set only sets one idx);
    // noobj zeroing: idempotent -> dedup.
    if (threadIdx.x == 0) {
        int nc = 0, nz = 0;
        for (int tt = 0; tt < NT; tt++) {
            if (!s_ok[tt]) continue;
            int key = (s_bn[tt] << 16) | (s_gj[tt] << 8) | s_gi[tt];
            int e = -1;
            for (int q = 0; q < nc; q++) if (s_ckey[q] == key) { e = q; break; }
            if (e < 0) {
                e = nc++; s_ckey[e] = key;
                s_cmask[e][0] = s_cmask[e][1] = s_cmask[e][2] = 0u;
            }
            s_ctb[e][0] = s_tb[tt][0]; s_ctb[e][1] = s_tb[tt][1];
            s_ctb[e][2] = s_tb[tt][2]; s_ctb[e][3] = s_tb[tt][3];
            s_cmask[e][s_cid[tt] >> 5] |= (1u << (s_cid[tt] & 31));
            int ign = s_ign[tt];
            for (int a = 0; a < 3; a++) {
                if (ign & (1 << a)) {
                    int zk = (a << 16) | (s_gj[tt] << 8) | s_gi[tt];
                    bool found = false;
                    for (int q = 0; q < nz; q++) if (s_zkey[q] == zk) { found = true; break; }
                    if (!found) s_zkey[nz++] = zk;
                }
            }
        }
        s_nc = nc; s_nz = nz;
    }
    __syncthreads();

    // phase 3: parallel sparse loss evaluation (fixed work partition -> deterministic)
    const int nc = s_nc, nz = s_nz;
    const int total = nc * (NCLS + 1) + nz;
    float acc = 0.0f;
    for (int i = threadIdx.x; i < total; i += 256) {
        if (i < nc * (NCLS + 1)) {
            int e = i / (NCLS + 1);
            int j = i - e * (NCLS + 1);
            int key = s_ckey[e];
            int bn = key >> 16, gj = (key >> 8) & 255, gi = key & 255;
            size_t base = (size_t)(b * 255 + bn * 85) * (size_t)HW + (size_t)gj * W + gi;
            if (j == 0) {
                // box (lambda_coord=0.5) + obj (lambda_obj=5)
                float px = sigmoidf_(P[base]);
                float py = sigmoidf_(P[base + (size_t)HW]);
                float pw = expf(P[base + 2 * (size_t)HW]);
                float ph = expf(P[base + 3 * (size_t)HW]);
                float pc = sigmoidf_(P[base + 4 * (size_t)HW]);
                float ci = ciou_xywh_dev(px, py, pw, ph,
                                         s_ctb[e][0], s_ctb[e][1], s_ctb[e][2], s_ctb[e][3]);
                acc += 0.5f * (1.0f - ci) + 5.0f * (pc - 1.0f) * (pc - 1.0f);
            } else {
                // class BCE (lambda_class=1)
                int c = j - 1;
                float p = sigmoidf_(P[base + (size_t)(5 + c) * (size_t)HW]);
                float lp = logf(p);      lp = lp < -100.0f ? -100.0f : lp;
                float ln = log1pf(-p);   ln = ln < -100.0f ? -100.0f : ln;
                int tbit = (s_cmask[e][c >> 5] >> (c & 31)) & 1;
                acc += -(tbit ? lp : ln);
            }
        } else {
            // noobj correction: subtract pconf^2 at deduped ignore cells (lambda_noobj=1)
            int zk = s_zkey[i - nc * (NCLS + 1)];
            int a = zk >> 16, gj = (zk >> 8) & 255, gi = zk & 255;
            float pc = sigmoidf_(P[(size_t)(b * 255 + a * 85 + 4) * (size_t)HW
                                   + (size_t)gj * W + gi]);
            acc -= pc * pc;
        }
    }
    red[threadIdx.x] = acc;
    __syncthreads();
    for (int s = 128; s > 0; s >>= 1) {
        if (threadIdx.x < s) red[threadIdx.x] += red[threadIdx.x + s];
        __syncthreads();
    }
    if (threadIdx.x == 0) partials[SLOT_CELL + blockIdx.x] = red[0];
}

// Single-wave final reduction of the 160 block partials using WMMA f32 16x16x4:
// A = 16x4 tile of partials (documented A layout: lanes 0-15 hold K=0/1 in the
// two A VGPRs, lanes 16-31 hold K=2/3), B = ones => D rows are chunk sums,
// C chains the accumulator across iterations. fp32-exact, one instr per 64 values.
__global__ __launch_bounds__(32)
void final_reduce_kernel(const float* __restrict__ part, float* __restrict__ out) {
    int lane = threadIdx.x;
    float total;
#if __has_builtin(__builtin_amdgcn_wmma_f32_16x16x4_f32)
    v8f c = {};
    for (int base = 0; base < NPART; base += 64) {
        int m    = lane & 15;
        int koff = (lane >> 4) * 2;       // lanes 16-31 hold K=2,3
        int i0 = base + m * 4 + koff;
        int i1 = i0 + 1;
        float a0 = part[i0 < NPART ? i0 : 0] * (i0 < NPART ? 1.0f : 0.0f);
        float a1 = part[i1 < NPART ? i1 : 0] * (i1 < NPART ? 1.0f : 0.0f);
        v2f a; a.x = a0; a.y = a1;
        v2f bo; bo.x = 1.0f; bo.y = 1.0f;
        c = __builtin_amdgcn_wmma_f32_16x16x4_f32(false, a, false, bo,
                                                  (short)0, c, false, false);
    }
    // D[m][n] identical over n; column 0: lane 0 holds M=0..7, lane 16 holds M=8..15
    float s = c[0] + c[1] + c[2] + c[3] + c[4] + c[5] + c[6] + c[7];
    float s16 = __shfl(s, 16, 32);
    total = s + s16;
#else
    total = 0.0f;
    if (lane == 0) for (int i = 0; i < NPART; i++) total += part[i];
#endif
    if (lane == 0) out[0] = total / (float)BATCH;
}

extern "C" void kernel_launch(void* const* d_in, const int* in_sizes, int n_in,
                              void* d_out, int out_size, void* d_ws, size_t ws_size,
                              hipStream_t stream) {
    const float* p3     = (const float*)d_in[0];
    const float* p4     = (const float*)d_in[1];
    const float* p5     = (const float*)d_in[2];
    const float* target = (const float*)d_in[3];
    float* part = (float*)d_ws;   // 160 floats; every slot written each call

    conf_sq_sum_kernel<<<64, 256, 0, stream>>>(p3, 76 * 76, BATCH * 3 * 76 * 76, part, SLOT_P3);
    conf_sq_sum_kernel<<<32, 256, 0, stream>>>(p4, 38 * 38, BATCH * 3 * 38 * 38, part, SLOT_P4);
    conf_sq_sum_kernel<<<16, 256, 0, stream>>>(p5, 19 * 19, BATCH * 3 * 19 * 19, part, SLOT_P5);
    cell_loss_kernel<<<3 * BATCH, 256, 0, stream>>>(p3, p4, p5, target, part);
    final_reduce_kernel<<<1, 32, 0, stream>>>(part, (float*)d_out);
}